// MPNN_Loop_40080634806381
// MI455X (gfx1250) — compile-verified
//
#include <hip/hip_runtime.h>

#define H 32
#define K2_WAVES 4
#define K2_TB (K2_WAVES * 32)

typedef __attribute__((ext_vector_type(2))) float v2f;
typedef __attribute__((ext_vector_type(8))) float v8f;

__device__ __forceinline__ v8f v8_bcast(float x) {
  v8f r;
#pragma unroll
  for (int i = 0; i < 8; ++i) r[i] = x;
  return r;
}

__device__ __forceinline__ v8f v8_lrelu(v8f x) {
#pragma unroll
  for (int i = 0; i < 8; ++i) {
    float v = x[i];
    x[i] = v > 0.0f ? v : 0.01f * v;
  }
  return x;
}

// Swizzled weight index: W is logically K x 32 row-major (row r, col c).
// A lane's B fragment for chunk k is {W[4k+2h][c], W[4k+2h+1][c]} -> store
// those two adjacently:  swz(r, c) = (r>>1)*64 + c*2 + (r&1).
__device__ __forceinline__ int wswz(int r, int c) {
  return ((r >> 1) << 6) + (c << 1) + (r & 1);
}

// One 16x16 output tile of D = A(16xK) x W(K x 32, one 16-col tile) + C using
// chained V_WMMA_F32_16X16X4_F32.
//   Arow = pointer to this lane's A row (row = lane % 16), contiguous K floats
//   Wswz = weights in swizzled (B-fragment contiguous) LDS layout
//   col  = (lane % 16) + 16*ctile ; half = lane / 16
__device__ __forceinline__ v8f gemm_rowmajor(const float* Arow,
                                             const float* Wswz, int K, int col,
                                             int half, v8f acc) {
#pragma unroll
  for (int k = 0; k < K; k += 4) {
    const int ka = k + 2 * half;
    v2f a = *(const v2f*)(Arow + ka);
    v2f b = *(const v2f*)(Wswz + ((k >> 1) + half) * 64 + col * 2);
    acc = __builtin_amdgcn_wmma_f32_16x16x4_f32(false, a, false, b, (short)0,
                                                acc, false, false);
  }
  return acc;
}

// ---------------------------------------------------------------- kernel 1
__global__ void mpnn_node_in(const float* __restrict__ x,
                             const float* __restrict__ W_in,
                             const float* __restrict__ b_in,
                             float* __restrict__ h_node,
                             float* __restrict__ aggr_out, long total) {
  long t = (long)blockIdx.x * blockDim.x + threadIdx.x;
  if (t >= total) return;
  int h = (int)(t & (H - 1));
  long node = t >> 5;
  float v = b_in[h] + x[node * 3 + 0] * W_in[0 * H + h] +
            x[node * 3 + 1] * W_in[1 * H + h] +
            x[node * 3 + 2] * W_in[2 * H + h];
  h_node[t] = v;
  aggr_out[t] = 0.0f;
}

// ---------------------------------------------------------------- kernel 2
__global__ __launch_bounds__(K2_TB) void mpnn_edge(
    const float* __restrict__ h_msg, const int* __restrict__ src,
    const int* __restrict__ dst, const float* __restrict__ h_node,
    const float* __restrict__ W_enc, const float* __restrict__ b_enc,
    const float* __restrict__ M_W1, const float* __restrict__ M_b1,
    const float* __restrict__ M_W2, const float* __restrict__ M_b2,
    const float* __restrict__ N_W1, const float* __restrict__ N_b1,
    const float* __restrict__ N_W2, const float* __restrict__ N_b2,
    const float* __restrict__ dec_W, const float* __restrict__ dec_b,
    float* __restrict__ out_aggr, float* __restrict__ out_ymsg,
    float* __restrict__ aggr_out, long n_edges) {
  __shared__ float sWenc[H * H];
  __shared__ float sMW1[3 * H * H];
  __shared__ float sMW2[H * H];
  __shared__ float sNW1[H * H];
  __shared__ float sNW2[H * H];
  __shared__ float sB[5 * H + 2 * H + 2];     // biases, dec_W(64), dec_b(2)
  __shared__ float sFeat[K2_WAVES][16 * 96];  // [h_i | h_j | encoded]
  __shared__ float sTmp[K2_WAVES][16 * H];    // A-relayout staging

  // cooperative weight load with B-fragment swizzle
  for (int i = threadIdx.x; i < H * H; i += K2_TB) {
    const int r = i >> 5, c = i & 31;
    const int s = wswz(r, c);
    sWenc[s] = W_enc[i];
    sMW2[s] = M_W2[i];
    sNW1[s] = N_W1[i];
    sNW2[s] = N_W2[i];
  }
  for (int i = threadIdx.x; i < 3 * H * H; i += K2_TB) {
    const int r = i >> 5, c = i & 31;
    sMW1[wswz(r, c)] = M_W1[i];
  }
  for (int i = threadIdx.x; i < H; i += K2_TB) {
    sB[i] = b_enc[i];
    sB[H + i] = M_b1[i];
    sB[2 * H + i] = M_b2[i];
    sB[3 * H + i] = N_b1[i];
    sB[4 * H + i] = N_b2[i];
  }
  for (int i = threadIdx.x; i < 2 * H; i += K2_TB) sB[5 * H + i] = dec_W[i];
  if (threadIdx.x < 2) sB[7 * H + threadIdx.x] = dec_b[threadIdx.x];
  __syncthreads();

  const int lane = threadIdx.x & 31;
  const int w = threadIdx.x >> 5;
  const int row = lane & 15;   // this lane's A row / B column (mod 16)
  const int half = lane >> 4;  // K sub-pair select

  const long e0 = ((long)blockIdx.x * K2_WAVES + w) * 16;
  const long erow = e0 + row;
  const bool rowOK = erow < n_edges;
  const long erowC = rowOK ? erow : (n_edges - 1);
  const int di = dst[erowC];
  const int si = src[erowC];

  float* F = &sFeat[w][0];
  float* T = &sTmp[w][0];

  // ---- gather h_i (=h_node[dst]) and h_j (=h_node[src]) into LDS ----------
  {
    const float* hp = &h_node[(size_t)di * H + 16 * half];
    float* fp = &F[row * 96 + 16 * half];
#pragma unroll
    for (int c = 0; c < 16; c += 4) *(float4*)(fp + c) = *(const float4*)(hp + c);
  }
  {
    const float* hp = &h_node[(size_t)si * H + 16 * half];
    float* fp = &F[row * 96 + 32 + 16 * half];
#pragma unroll
    for (int c = 0; c < 16; c += 4) *(float4*)(fp + c) = *(const float4*)(hp + c);
  }

  // ---- encoded = h_msg @ W_enc + b_enc (A fragments straight from global) --
  {
    const float* mrow = &h_msg[(size_t)erowC * H];
#pragma unroll
    for (int ct = 0; ct < 2; ++ct) {
      const int col = row + 16 * ct;
      v8f acc = v8_bcast(sB[col]);
      acc = gemm_rowmajor(mrow, sWenc, H, col, half, acc);
#pragma unroll
      for (int r = 0; r < 8; ++r) F[(half * 8 + r) * 96 + 64 + col] = acc[r];
    }
  }
  __syncthreads();

  // ---- N branch: n = lrelu(lrelu(h_j @ N_W1 + b) @ N_W2 + b) ---------------
  v8f nD[2];
#pragma unroll
  for (int ct = 0; ct < 2; ++ct) {
    const int col = row + 16 * ct;
    v8f acc = v8_bcast(sB[3 * H + col]);
    acc = gemm_rowmajor(&F[row * 96 + 32], sNW1, H, col, half, acc);
    acc = v8_lrelu(acc);
#pragma unroll
    for (int r = 0; r < 8; ++r) T[(half * 8 + r) * H + col] = acc[r];
  }
  __syncthreads();
#pragma unroll
  for (int ct = 0; ct < 2; ++ct) {
    const int col = row + 16 * ct;
    v8f acc = v8_bcast(sB[4 * H + col]);
    acc = gemm_rowmajor(&T[row * H], sNW2, H, col, half, acc);
    nD[ct] = v8_lrelu(acc);
  }
  __syncthreads();

  // ---- M branch: m = lrelu(lrelu([h_i|h_j|enc] @ M_W1 + b) @ M_W2 + b) ----
#pragma unroll
  for (int ct = 0; ct < 2; ++ct) {
    const int col = row + 16 * ct;
    v8f acc = v8_bcast(sB[H + col]);
    acc = gemm_rowmajor(&F[row * 96], sMW1, 3 * H, col, half, acc);
    acc = v8_lrelu(acc);
#pragma unroll
    for (int r = 0; r < 8; ++r) T[(half * 8 + r) * H + col] = acc[r];
  }
  __syncthreads();
  v8f aggr[2];
#pragma unroll
  for (int ct = 0; ct < 2; ++ct) {
    const int col = row + 16 * ct;
    v8f acc = v8_bcast(sB[2 * H + col]);
    acc = gemm_rowmajor(&T[row * H], sMW2, H, col, half, acc);
    acc = v8_lrelu(acc);
#pragma unroll
    for (int r = 0; r < 8; ++r) aggr[ct][r] = acc[r] + nD[ct][r];
  }
  __syncthreads();  // T reads (M2) done before re-staging aggr below

  // ---- store aggr_msg (global + LDS restage) -------------------------------
#pragma unroll
  for (int ct = 0; ct < 2; ++ct) {
    const int col = row + 16 * ct;
#pragma unroll
    for (int r = 0; r < 8; ++r) {
      const long e = e0 + half * 8 + r;
      T[(half * 8 + r) * H + col] = aggr[ct][r];
      if (e < n_edges) out_aggr[(size_t)e * H + col] = aggr[ct][r];
    }
  }
  __syncthreads();

  // ---- y_msg = softmax(aggr @ dec_W + dec_b) -------------------------------
  if (lane < 16) {
    const long e = e0 + lane;
    if (e < n_edges) {
      const float* a = &T[lane * H];
      float z0 = sB[7 * H + 0], z1 = sB[7 * H + 1];
#pragma unroll
      for (int h = 0; h < H; ++h) {
        const float v = a[h];
        z0 += v * sB[5 * H + h * 2 + 0];
        z1 += v * sB[5 * H + h * 2 + 1];
      }
      const float mx = fmaxf(z0, z1);
      const float p0 = __expf(z0 - mx), p1 = __expf(z1 - mx);
      const float inv = 1.0f / (p0 + p1);
      out_ymsg[e * 2 + 0] = p0 * inv;
      out_ymsg[e * 2 + 1] = p1 * inv;
    }
  }

  // ---- segment_sum over dst via f32 atomics --------------------------------
  if (rowOK) {
    float* ap = &aggr_out[(size_t)di * H + 16 * half];
    const float* sp = &T[row * H + 16 * half];
#pragma unroll
    for (int c = 0; c < 16; ++c) atomicAdd(ap + c, sp[c]);
  }
}

// ---------------------------------------------------------------- kernel 3
__global__ __launch_bounds__(128) void mpnn_belief(
    const float* __restrict__ h_node, const float* __restrict__ aggr_out,
    const float* __restrict__ U_W, const float* __restrict__ U_b,
    const float* __restrict__ bel_W, const float* __restrict__ bel_b,
    float* __restrict__ out_bel, int n_var) {
  __shared__ float sU[2 * H * H];
  __shared__ float sUb[H];
  __shared__ float sbW[2 * H];
  __shared__ float sbb[2];
  for (int i = threadIdx.x; i < 2 * H * H; i += 128) sU[i] = U_W[i];
  for (int i = threadIdx.x; i < H; i += 128) sUb[i] = U_b[i];
  for (int i = threadIdx.x; i < 2 * H; i += 128) sbW[i] = bel_W[i];
  if (threadIdx.x < 2) sbb[threadIdx.x] = bel_b[threadIdx.x];
  __syncthreads();

  const long i = (long)blockIdx.x * blockDim.x + threadIdx.x;
  if (i >= n_var) return;

  float u[H];
#pragma unroll
  for (int j = 0; j < H; ++j) u[j] = sUb[j];
  const float* hn = &h_node[(size_t)i * H];
  const float* ag = &aggr_out[(size_t)i * H];
  for (int k = 0; k < H; ++k) {
    const float v = hn[k];
#pragma unroll
    for (int j = 0; j < H; ++j) u[j] += v * sU[k * H + j];
  }
  for (int k = 0; k < H; ++k) {
    const float v = ag[k];
#pragma unroll
    for (int j = 0; j < H; ++j) u[j] += v * sU[(H + k) * H + j];
  }
  float z0 = sbb[0], z1 = sbb[1];
#pragma unroll
  for (int j = 0; j < H; ++j) {
    const float v = u[j] > 0.0f ? u[j] : 0.01f * u[j];
    z0 += v * sbW[j * 2 + 0];
    z1 += v * sbW[j * 2 + 1];
  }
  const float mx = fmaxf(z0, z1);
  const float p0 = __expf(z0 - mx), p1 = __expf(z1 - mx);
  const float inv = 1.0f / (p0 + p1);
  out_bel[i * 2 + 0] = p0 * inv;
  out_bel[i * 2 + 1] = p1 * inv;
}

// ---------------------------------------------------------------- launcher
extern "C" void kernel_launch(void* const* d_in, const int* in_sizes, int n_in,
                              void* d_out, int out_size, void* d_ws,
                              size_t ws_size, hipStream_t stream) {
  (void)n_in;
  (void)ws_size;
  const float* x = (const float*)d_in[0];
  const float* h_msg = (const float*)d_in[1];
  const int* eidx = (const int*)d_in[2];
  const float* W_in = (const float*)d_in[4];
  const float* b_in = (const float*)d_in[5];
  const float* W_enc = (const float*)d_in[6];
  const float* b_enc = (const float*)d_in[7];
  const float* M_W1 = (const float*)d_in[8];
  const float* M_b1 = (const float*)d_in[9];
  const float* M_W2 = (const float*)d_in[10];
  const float* M_b2 = (const float*)d_in[11];
  const float* N_W1 = (const float*)d_in[12];
  const float* N_b1 = (const float*)d_in[13];
  const float* N_W2 = (const float*)d_in[14];
  const float* N_b2 = (const float*)d_in[15];
  const float* U_W = (const float*)d_in[16];
  const float* U_b = (const float*)d_in[17];
  const float* dec_W = (const float*)d_in[18];
  const float* dec_b = (const float*)d_in[19];
  const float* bel_W = (const float*)d_in[20];
  const float* bel_b = (const float*)d_in[21];

  const long n_nodes = in_sizes[0] / 3;
  const long n_edges = in_sizes[1] / H;
  const int n_var = (int)(((long)out_size - n_edges * (H + 2)) / 2);

  float* out_aggr = (float*)d_out;
  float* out_ymsg = out_aggr + (size_t)n_edges * H;
  float* out_bel = out_ymsg + (size_t)n_edges * 2;

  float* h_node = (float*)d_ws;
  float* aggr_out = h_node + (size_t)n_nodes * H;

  const int* src = eidx;
  const int* dst = eidx + n_edges;

  const long t1 = n_nodes * H;
  mpnn_node_in<<<dim3((unsigned)((t1 + 255) / 256)), 256, 0, stream>>>(
      x, W_in, b_in, h_node, aggr_out, t1);

  const long ntiles = (n_edges + 15) / 16;
  const long nblk = (ntiles + K2_WAVES - 1) / K2_WAVES;
  mpnn_edge<<<dim3((unsigned)nblk), K2_TB, 0, stream>>>(
      h_msg, src, dst, h_node, W_enc, b_enc, M_W1, M_b1, M_W2, M_b2, N_W1,
      N_b1, N_W2, N_b2, dec_W, dec_b, out_aggr, out_ymsg, aggr_out, n_edges);

  mpnn_belief<<<dim3((unsigned)((n_var + 127) / 128)), 128, 0, stream>>>(
      h_node, aggr_out, U_W, U_b, bel_W, bel_b, out_bel, n_var);
}